// ScalingSquaring_6966436954312
// MI455X (gfx1250) — compile-verified
//
#include <hip/hip_runtime.h>
#include <math.h>

// Shape fixed by the reference: velocity (B=2, C=3, D=H=W=128), fp32.
static constexpr unsigned DHW   = 128u * 128u * 128u;   // 2^21 voxels per batch
static constexpr unsigned NVOX  = 2u * DHW;             // both batches
static constexpr float    INV64 = 1.0f / 64.0f;         // 2^-NUM_SQUARING

// ---------------- small float4 helpers (lower to v_pk_* f32) ----------------
__device__ __forceinline__ float4 f4sub(const float4& a, const float4& b) {
    return make_float4(a.x - b.x, a.y - b.y, a.z - b.z, a.w - b.w);
}
__device__ __forceinline__ float4 f4fma(float s, const float4& a, const float4& b) {
    // b + s*a, per component
    return make_float4(fmaf(s, a.x, b.x), fmaf(s, a.y, b.y),
                       fmaf(s, a.z, b.z), fmaf(s, a.w, b.w));
}
__device__ __forceinline__ float4 f4lerp(float w, const float4& v0, const float4& v1) {
    return f4fma(w, f4sub(v1, v0), v0);
}

// ---------------------------------------------------------------------------
// Sample position -> 8 corner offsets + weights. Matches reference semantics:
// sample=(x,y,z)+flow ; nan->0 ; *2/127-1 ; clip(+-0.999) ; unnormalize ;
// border-clamped trilinear.
// ---------------------------------------------------------------------------
__device__ __forceinline__ void corner_setup(unsigned x, unsigned y, unsigned z,
                                             float f2x, float f2y, float f2z,
                                             unsigned idx[8],
                                             float& wx, float& wy, float& wz) {
    float sx = (float)x + f2x;  sx = (sx == sx) ? sx : 0.0f;
    float sy = (float)y + f2y;  sy = (sy == sy) ? sy : 0.0f;
    float sz = (float)z + f2z;  sz = (sz == sz) ? sz : 0.0f;

    const float nrm = 2.0f / 127.0f;
    float ux = fminf(fmaxf(fmaf(sx, nrm, -1.0f), -0.999f), 0.999f);
    float uy = fminf(fmaxf(fmaf(sy, nrm, -1.0f), -0.999f), 0.999f);
    float uz = fminf(fmaxf(fmaf(sz, nrm, -1.0f), -0.999f), 0.999f);
    float X = fmaf(ux, 63.5f, 63.5f);
    float Y = fmaf(uy, 63.5f, 63.5f);
    float Z = fmaf(uz, 63.5f, 63.5f);

    float xf = floorf(X), yf = floorf(Y), zf = floorf(Z);
    wx = X - xf;  wy = Y - yf;  wz = Z - zf;

    int x0 = min(max((int)xf, 0), 127);
    int y0 = min(max((int)yf, 0), 127);
    int z0 = min(max((int)zf, 0), 127);
    int x1 = min(x0 + 1, 127);
    int y1 = min(y0 + 1, 127);
    int z1 = min(z0 + 1, 127);

    unsigned zo0 = (unsigned)z0 << 14, zo1 = (unsigned)z1 << 14;
    unsigned yo0 = (unsigned)y0 << 7,  yo1 = (unsigned)y1 << 7;
    idx[0] = zo0 + yo0 + (unsigned)x0;
    idx[1] = zo0 + yo0 + (unsigned)x1;
    idx[2] = zo0 + yo1 + (unsigned)x0;
    idx[3] = zo0 + yo1 + (unsigned)x1;
    idx[4] = zo1 + yo0 + (unsigned)x0;
    idx[5] = zo1 + yo0 + (unsigned)x1;
    idx[6] = zo1 + yo1 + (unsigned)x0;
    idx[7] = zo1 + yo1 + (unsigned)x1;
}

// Core of one squaring step on AoS input: returns updated flow (x,y,z,0).
__device__ __forceinline__ float4 compose_aos_body(const float4* __restrict__ src,
                                                   unsigned t) {
    unsigned vox  = t & (DHW - 1u);
    unsigned bofs = t & ~(DHW - 1u);          // b * DHW
    unsigned x =  vox         & 127u;
    unsigned y = (vox >> 7)   & 127u;
    unsigned z = (vox >> 14)  & 127u;

    float4 f2 = src[t];                        // flow2 at own voxel (b128)

    unsigned idx[8];
    float wx, wy, wz;
    corner_setup(x, y, z, f2.x, f2.y, f2.z, idx, wx, wy, wz);

    const float4* __restrict__ base = src + bofs;
    float4 v[8];
#pragma unroll
    for (int k = 0; k < 8; ++k) v[k] = base[idx[k]];   // 8x global_load_b128

    float4 c00 = f4lerp(wx, v[0], v[1]);
    float4 c01 = f4lerp(wx, v[2], v[3]);
    float4 c10 = f4lerp(wx, v[4], v[5]);
    float4 c11 = f4lerp(wx, v[6], v[7]);
    float4 c0  = f4lerp(wy, c00, c01);
    float4 c1  = f4lerp(wy, c10, c11);
    float4 r   = f4lerp(wz, c0, c1);

    r.x = (r.x == r.x) ? r.x : 0.0f;           // nan_to_num on sampled value
    r.y = (r.y == r.y) ? r.y : 0.0f;
    r.z = (r.z == r.z) ? r.z : 0.0f;

    return make_float4(r.x + f2.x, r.y + f2.y, r.z + f2.z, 0.0f);
}

// ===========================================================================
// AoS path (preferred): flow stored as float4 (x,y,z,0), 16B aligned.
// ===========================================================================

// planar velocity -> AoS, fused with *2^-6
__global__ __launch_bounds__(256) void pack_kernel(const float* __restrict__ vel,
                                                   float4* __restrict__ dst) {
    unsigned t   = blockIdx.x * 256u + threadIdx.x;
    unsigned vox = t & (DHW - 1u);
    unsigned b   = t >> 21;
    const float* __restrict__ p = vel + (size_t)b * (3u * DHW);
    dst[t] = make_float4(p[vox] * INV64,
                         p[DHW + vox] * INV64,
                         p[2u * DHW + vox] * INV64,
                         0.0f);
}

// Middle squaring step: AoS -> AoS (1 b128 load + 8 b128 gathers + 1 b128 store)
__global__ __launch_bounds__(256) void compose_aos_kernel(const float4* __restrict__ src,
                                                          float4* __restrict__ dst) {
    unsigned t = blockIdx.x * 256u + threadIdx.x;
    dst[t] = compose_aos_body(src, t);
}

// Final squaring step fused with unpack: AoS -> planar d_out
// (1 b128 load + 8 b128 gathers + 3 b32 stores; saves a full volume pass)
__global__ __launch_bounds__(256) void compose_aos_out_kernel(const float4* __restrict__ src,
                                                              float* __restrict__ out) {
    unsigned t   = blockIdx.x * 256u + threadIdx.x;
    unsigned vox = t & (DHW - 1u);
    unsigned b   = t >> 21;

    float4 o = compose_aos_body(src, t);

    float* __restrict__ q = out + (size_t)b * (3u * DHW);
    q[vox]            = o.x;
    q[DHW + vox]      = o.y;
    q[2u * DHW + vox] = o.z;
}

// ===========================================================================
// Planar fallback path (workspace only needs one 50 MB buffer).
// ===========================================================================
__global__ __launch_bounds__(256) void vel_scale_kernel(const float4* __restrict__ v,
                                                        float4* __restrict__ o,
                                                        unsigned n4) {
    unsigned i = blockIdx.x * 256u + threadIdx.x;
    if (i < n4) {
        float4 a = v[i];
        o[i] = make_float4(a.x * INV64, a.y * INV64, a.z * INV64, a.w * INV64);
    }
}

__global__ __launch_bounds__(256) void compose_kernel(const float* __restrict__ src,
                                                      float* __restrict__ dst) {
    unsigned t   = blockIdx.x * 256u + threadIdx.x;
    unsigned vox = t & (DHW - 1u);
    unsigned b   = t >> 21;
    unsigned x =  vox         & 127u;
    unsigned y = (vox >> 7)   & 127u;
    unsigned z = (vox >> 14)  & 127u;

    const float* __restrict__ p0 = src + (size_t)b * (3u * DHW);
    const float* __restrict__ p1 = p0 + DHW;
    const float* __restrict__ p2 = p1 + DHW;

    float f2x = p0[vox], f2y = p1[vox], f2z = p2[vox];

    unsigned idx[8];
    float wx, wy, wz;
    corner_setup(x, y, z, f2x, f2y, f2z, idx, wx, wy, wz);

    float vx[8], vy[8], vz[8];
#pragma unroll
    for (int k = 0; k < 8; ++k) vx[k] = p0[idx[k]];
#pragma unroll
    for (int k = 0; k < 8; ++k) vy[k] = p1[idx[k]];
#pragma unroll
    for (int k = 0; k < 8; ++k) vz[k] = p2[idx[k]];

    auto tri = [&](const float v[8]) -> float {
        float c00 = fmaf(wx, v[1] - v[0], v[0]);
        float c01 = fmaf(wx, v[3] - v[2], v[2]);
        float c10 = fmaf(wx, v[5] - v[4], v[4]);
        float c11 = fmaf(wx, v[7] - v[6], v[6]);
        float c0  = fmaf(wy, c01 - c00, c00);
        float c1  = fmaf(wy, c11 - c10, c10);
        return fmaf(wz, c1 - c0, c0);
    };

    float rx = tri(vx);  rx = (rx == rx) ? rx : 0.0f;
    float ry = tri(vy);  ry = (ry == ry) ? ry : 0.0f;
    float rz = tri(vz);  rz = (rz == rz) ? rz : 0.0f;

    float* __restrict__ q0 = dst + (size_t)b * (3u * DHW);
    q0[vox]            = rx + f2x;
    q0[DHW + vox]      = ry + f2y;
    q0[2u * DHW + vox] = rz + f2z;
}

// ---------------------------------------------------------------------------
// Launch. AoS path if workspace holds two float4 volumes (134 MB), else
// planar ping-pong d_out <-> d_ws. Branch depends only on ws_size (constant
// across calls -> deterministic). All launches on `stream`, capture-safe.
// ---------------------------------------------------------------------------
extern "C" void kernel_launch(void* const* d_in, const int* in_sizes, int n_in,
                              void* d_out, int out_size, void* d_ws, size_t ws_size,
                              hipStream_t stream) {
    const float* vel = (const float*)d_in[0];
    float* out = (float*)d_out;

    const dim3 blk(256);
    const dim3 grid(NVOX / 256u);              // 16384 blocks, 8 wave32 waves each

    const size_t aosBuf = (size_t)NVOX * sizeof(float4);   // 67,108,864 bytes

    if (ws_size >= 2 * aosBuf) {
        // ---------------- AoS pipeline: pack, 5x compose, fused final ----------------
        float4* wsA = (float4*)d_ws;
        float4* wsB = wsA + NVOX;

        pack_kernel<<<grid, blk, 0, stream>>>(vel, wsA);

        float4* a = wsA;                       // source of next compose
        float4* b = wsB;
        for (int it = 0; it < 5; ++it) {       // squarings 1..5: A->B->A->B->A->B
            compose_aos_kernel<<<grid, blk, 0, stream>>>(a, b);
            float4* tmp = a; a = b; b = tmp;
        }
        // after 5 swaps a == wsB holds the 5x-composed flow; squaring 6 fused
        // with AoS->planar unpack, writing d_out directly.
        compose_aos_out_kernel<<<grid, blk, 0, stream>>>(a, out);
    } else {
        // ---------------- planar fallback ----------------
        float* ws = (float*)d_ws;
        const unsigned n  = (unsigned)in_sizes[0];   // 12,582,912 floats
        const unsigned n4 = n / 4u;

        vel_scale_kernel<<<(n4 + 255u) / 256u, blk, 0, stream>>>(
            (const float4*)vel, (float4*)out, n4);

        float* a = out;
        float* b = ws;
        for (int it = 0; it < 6; ++it) {       // even count: result lands in d_out
            compose_kernel<<<grid, blk, 0, stream>>>(a, b);
            float* tmp = a; a = b; b = tmp;
        }
    }
}